// GINLayer_78589311582938
// MI455X (gfx1250) — compile-verified
//
#include <hip/hip_runtime.h>

#define DD 64
#define WPITCH 65      // LDS pitch for 64x64 weight tiles (bank-conflict pad)
#define XPITCH 68      // LDS pitch for 16x64 activation tiles (bank-conflict pad)
#define BN_EPS 1e-5f

typedef __attribute__((ext_vector_type(2))) float v2f;
typedef __attribute__((ext_vector_type(8))) float v8f;

// ---------------------------------------------------------------- zero ws
__global__ void gin_zero_kernel(float* __restrict__ p, long n) {
    long i = (long)blockIdx.x * blockDim.x + threadIdx.x;
    long stride = (long)gridDim.x * blockDim.x;
    for (; i < n; i += stride) p[i] = 0.0f;
}

// ------------------------------------------------------- scatter-sum h[src] -> neigh[dst]
// one thread = one edge x 4 features; neigh (25.6MB) is L2-resident so the
// f32 atomics resolve on-chip.
__global__ void gin_scatter_kernel(const float* __restrict__ h,
                                   const int* __restrict__ src,
                                   const int* __restrict__ dst,
                                   float* __restrict__ neigh, int nE) {
    long tid = (long)blockIdx.x * blockDim.x + threadIdx.x;
    int e = (int)(tid >> 4);
    if (e >= nE) return;
    int c = (int)(tid & 15) << 2;
    int s = src[e];
    int d = dst[e];
    float4 v = *(const float4*)(h + (size_t)s * DD + c);
    float* p = neigh + (size_t)d * DD + c;
    atomicAdd(p + 0, v.x);
    atomicAdd(p + 1, v.y);
    atomicAdd(p + 2, v.z);
    atomicAdd(p + 3, v.w);
}

// ------------------------------------------------------- fused 2-layer MLP + BN stats
// Block = 128 threads = 4 waves; each wave owns a 16-row tile.
// GEMMs use v_wmma_f32_16x16x4_f32:
//   A 16x4 layout: lanes 0-15 -> row=lane, vgpr0=K0 vgpr1=K1; lanes 16-31 -> K2,K3
//   B 4x16 layout: vgpr0: K=k+2*half, vgpr1: K=k+2*half+1; col = lane&15
//   C/D 16x16:     lanes 0-15: vgpr v = row v; lanes 16-31: vgpr v = row v+8; col = lane&15
template <bool USE_NEIGH>
__global__ void __launch_bounds__(128)
gin_mlp_kernel(const float* __restrict__ Xin, const float* __restrict__ neigh,
               const float* __restrict__ eps_p,
               const float* __restrict__ W1, const float* __restrict__ b1,
               const float* __restrict__ W2, const float* __restrict__ b2,
               float* __restrict__ Y,
               float* __restrict__ sum_g, float* __restrict__ sumsq_g,
               int nrows) {
    __shared__ float W1s[DD * WPITCH];
    __shared__ float W2s[DD * WPITCH];
    __shared__ float Xs[4 * 16 * XPITCH];
    __shared__ float sSum[DD];
    __shared__ float sSq[DD];

    const int tid  = threadIdx.x;
    const int lane = tid & 31;
    const int wave = tid >> 5;
    const int half = lane >> 4;   // which 16-lane half
    const int l16  = lane & 15;

    if (tid < DD) { sSum[tid] = 0.0f; sSq[tid] = 0.0f; }

    // stage both weight matrices (row-major, padded pitch)
    for (int i = tid; i < DD * DD; i += 128) {
        int r = i >> 6, c = i & 63;
        W1s[r * WPITCH + c] = W1[i];
        W2s[r * WPITCH + c] = W2[i];
    }

    const int row_base = (blockIdx.x * 4 + wave) * 16;
    const float scale = USE_NEIGH ? (1.0f + eps_p[0]) : 1.0f;

    // stage this wave's 16x64 input tile: x = (1+eps)*h + neigh  (or just e)
    float* Xw = &Xs[wave * 16 * XPITCH];
    {
        int lrow = lane >> 1;
        int c0 = (lane & 1) * 32;
        int grow = row_base + lrow;
        for (int j = 0; j < 32; j += 4) {
            float4 v = make_float4(0.f, 0.f, 0.f, 0.f);
            if (grow < nrows) {
                float4 xv = *(const float4*)(Xin + (size_t)grow * DD + c0 + j);
                if (USE_NEIGH) {
                    float4 nv = *(const float4*)(neigh + (size_t)grow * DD + c0 + j);
                    v = make_float4(scale * xv.x + nv.x, scale * xv.y + nv.y,
                                    scale * xv.z + nv.z, scale * xv.w + nv.w);
                } else {
                    v = xv;
                }
            }
            float* p = &Xw[lrow * XPITCH + c0 + j];
            p[0] = v.x; p[1] = v.y; p[2] = v.z; p[3] = v.w;
        }
    }
    __syncthreads();

    float bias1[4], bias2[4];
#pragma unroll
    for (int nt = 0; nt < 4; ++nt) {
        bias1[nt] = b1[nt * 16 + l16];
        bias2[nt] = b2[nt * 16 + l16];
    }

    // ---------------- GEMM 1: acc = X @ W1
    v8f acc[4] = {{0,0,0,0,0,0,0,0}, {0,0,0,0,0,0,0,0},
                  {0,0,0,0,0,0,0,0}, {0,0,0,0,0,0,0,0}};
#pragma unroll
    for (int kk = 0; kk < 16; ++kk) {
        int k = kk * 4;
        v2f a = *(const v2f*)&Xw[l16 * XPITCH + k + 2 * half];
#pragma unroll
        for (int nt = 0; nt < 4; ++nt) {
            v2f b;
            b.x = W1s[(k + 2 * half + 0) * WPITCH + nt * 16 + l16];
            b.y = W1s[(k + 2 * half + 1) * WPITCH + nt * 16 + l16];
            acc[nt] = __builtin_amdgcn_wmma_f32_16x16x4_f32(
                false, a, false, b, (short)0, acc[nt], false, false);
        }
    }

    // y1 = relu(acc + b1) -> back into this wave's LDS tile (wave-private, in-order LDS)
#pragma unroll
    for (int nt = 0; nt < 4; ++nt) {
        int c = nt * 16 + l16;
#pragma unroll
        for (int v = 0; v < 8; ++v) {
            int m = v + 8 * half;
            float y = fmaxf(acc[nt][v] + bias1[nt], 0.0f);
            Xw[m * XPITCH + c] = y;
        }
    }
    __syncthreads();

    // ---------------- GEMM 2: acc2 = Y1 @ W2
    v8f acc2[4] = {{0,0,0,0,0,0,0,0}, {0,0,0,0,0,0,0,0},
                   {0,0,0,0,0,0,0,0}, {0,0,0,0,0,0,0,0}};
#pragma unroll
    for (int kk = 0; kk < 16; ++kk) {
        int k = kk * 4;
        v2f a = *(const v2f*)&Xw[l16 * XPITCH + k + 2 * half];
#pragma unroll
        for (int nt = 0; nt < 4; ++nt) {
            v2f b;
            b.x = W2s[(k + 2 * half + 0) * WPITCH + nt * 16 + l16];
            b.y = W2s[(k + 2 * half + 1) * WPITCH + nt * 16 + l16];
            acc2[nt] = __builtin_amdgcn_wmma_f32_16x16x4_f32(
                false, a, false, b, (short)0, acc2[nt], false, false);
        }
    }

    // y2 = acc2 + b2: store pre-norm output, accumulate per-column BN stats
#pragma unroll
    for (int nt = 0; nt < 4; ++nt) {
        int c = nt * 16 + l16;
        float s = 0.0f, q = 0.0f;
#pragma unroll
        for (int v = 0; v < 8; ++v) {
            int m = v + 8 * half;
            int grow = row_base + m;
            float y = acc2[nt][v] + bias2[nt];
            if (grow < nrows) {
                Y[(size_t)grow * DD + c] = y;
                s += y;
                q += y * y;
            }
        }
        atomicAdd(&sSum[c], s);   // ds_add_f32
        atomicAdd(&sSq[c], q);
    }
    __syncthreads();
    if (tid < DD) {
        atomicAdd(&sum_g[tid], sSum[tid]);
        atomicAdd(&sumsq_g[tid], sSq[tid]);
    }
}

// ------------------------------------------------------- fold stats -> per-column a,b
// out = relu(y*a + b) with a = gamma*rsqrt(var+eps), b = beta - mean*a
__global__ void gin_bn_params_kernel(const float* __restrict__ stats,
                                     const float* __restrict__ gamma_h,
                                     const float* __restrict__ beta_h,
                                     const float* __restrict__ gamma_e,
                                     const float* __restrict__ beta_e,
                                     float* __restrict__ params,
                                     float invN_h, float invN_e) {
    int t = threadIdx.x;
    if (t < DD) {
        float m = stats[t] * invN_h;
        float v = stats[DD + t] * invN_h - m * m;
        float a = gamma_h[t] * rsqrtf(v + BN_EPS);
        params[t] = a;
        params[DD + t] = beta_h[t] - m * a;
    } else if (t < 2 * DD) {
        int c = t - DD;
        float m = stats[2 * DD + c] * invN_e;
        float v = stats[3 * DD + c] * invN_e - m * m;
        float a = gamma_e[c] * rsqrtf(v + BN_EPS);
        params[2 * DD + c] = a;
        params[3 * DD + c] = beta_e[c] - m * a;
    }
}

// ------------------------------------------------------- normalize+relu+residual in place
__global__ void gin_finalize_kernel(float* __restrict__ Y,
                                    const float* __restrict__ Xres,
                                    const float* __restrict__ params, // a[64], b[64]
                                    long n4) {
    long i = (long)blockIdx.x * blockDim.x + threadIdx.x;
    if (i >= n4) return;
    float4 y = ((const float4*)Y)[i];
    float4 r = ((const float4*)Xres)[i];
    int c = (int)((i << 2) & 63);
    float a0 = params[c + 0], a1 = params[c + 1], a2 = params[c + 2], a3 = params[c + 3];
    float b0 = params[DD + c + 0], b1 = params[DD + c + 1];
    float b2 = params[DD + c + 2], b3 = params[DD + c + 3];
    y.x = r.x + fmaxf(y.x * a0 + b0, 0.0f);
    y.y = r.y + fmaxf(y.y * a1 + b1, 0.0f);
    y.z = r.z + fmaxf(y.z * a2 + b2, 0.0f);
    y.w = r.w + fmaxf(y.w * a3 + b3, 0.0f);
    ((float4*)Y)[i] = y;
}

// ----------------------------------------------------------------------------
extern "C" void kernel_launch(void* const* d_in, const int* in_sizes, int n_in,
                              void* d_out, int out_size, void* d_ws, size_t ws_size,
                              hipStream_t stream) {
    const float* h       = (const float*)d_in[0];
    const float* e       = (const float*)d_in[1];
    const int*   src     = (const int*)d_in[2];
    const int*   dst     = (const int*)d_in[3];
    const float* eps     = (const float*)d_in[4];
    const float* W1      = (const float*)d_in[5];
    const float* b1      = (const float*)d_in[6];
    const float* W2      = (const float*)d_in[7];
    const float* b2      = (const float*)d_in[8];
    const float* gamma_h = (const float*)d_in[9];
    const float* beta_h  = (const float*)d_in[10];
    const float* gamma_e = (const float*)d_in[11];
    const float* beta_e  = (const float*)d_in[12];

    const int N = in_sizes[0] / DD;   // 100000 nodes
    const int E = in_sizes[1] / DD;   // 1600000 edges

    float* ws     = (float*)d_ws;
    float* neigh  = ws;                         // N*DD floats
    float* stats  = ws + (size_t)N * DD;        // 256 floats: sum_h,sq_h,sum_e,sq_e
    float* params = stats + 4 * DD;             // 256 floats: a_h,b_h,a_e,b_e

    float* Yh = (float*)d_out;                  // node output region (pre-norm, then final)
    float* Ye = Yh + (size_t)N * DD;            // edge output region

    // 1. zero neigh accumulator + stats
    long nz = (long)N * DD + 4 * DD;
    gin_zero_kernel<<<2048, 256, 0, stream>>>(ws, nz);

    // 2. scatter-sum of h[src] onto dst
    long nscatter = (long)E * 16;
    gin_scatter_kernel<<<(int)((nscatter + 255) / 256), 256, 0, stream>>>(
        h, src, dst, neigh, E);

    // 3. fused MLP + BN-stat passes (node then edge; independent, stream-ordered)
    gin_mlp_kernel<true><<<(N + 63) / 64, 128, 0, stream>>>(
        h, neigh, eps, W1, b1, W2, b2, Yh, stats, stats + DD, N);
    gin_mlp_kernel<false><<<(E + 63) / 64, 128, 0, stream>>>(
        e, nullptr, eps, W1, b1, W2, b2, Ye, stats + 2 * DD, stats + 3 * DD, E);

    // 4. fold stats into per-column scale/shift
    gin_bn_params_kernel<<<1, 128, 0, stream>>>(
        stats, gamma_h, beta_h, gamma_e, beta_e, params, 1.0f / N, 1.0f / E);

    // 5. normalize + relu + residual, in place over d_out
    long n4h = (long)N * DD / 4;
    long n4e = (long)E * DD / 4;
    gin_finalize_kernel<<<(int)((n4h + 255) / 256), 256, 0, stream>>>(Yh, h, params, n4h);
    gin_finalize_kernel<<<(int)((n4e + 255) / 256), 256, 0, stream>>>(Ye, e, params + 2 * DD, n4e);
}